// ExponentialSmoothing_79852031967458
// MI455X (gfx1250) — compile-verified
//
#include <hip/hip_runtime.h>

// EMA scan y_t = 0.1*x_t + 0.9*y_{t-1} over x[8, 8192, 512] f32.
// Chunked parallel scan (warmup 256 => 0.9^256 ~ 2e-12, below fp32 eps).
// TDM (tensor_load_to_lds / TENSORcnt) feeds a 6-deep LDS ring, 5 tiles
// (160 KB) in flight per block; compute reads LDS b128, stores NT b128.

namespace {
constexpr int kB       = 8;
constexpr int kT       = 8192;
constexpr int kF       = 512;
constexpr int kChunk   = 512;   // timesteps stored per block
constexpr int kWarm    = 256;   // warmup steps for carry reconstruction
constexpr int kTT      = 16;    // timesteps per LDS tile (tile = 32 KB, contiguous)
constexpr int kThreads = 128;   // each thread owns 4 consecutive features
constexpr int kNBuf    = 6;     // LDS ring buffers (6 * 32 KB = 192 KB of 320 KB)
constexpr int kDepth   = 5;     // tiles in flight per block (TENSORcnt-tracked)
constexpr int kTileElems = kTT * kF;   // 8192 elements, fits 16-bit tile_dim0
constexpr float kAlpha = 0.1f;
constexpr float kBeta  = 0.9f;

typedef float f4 __attribute__((ext_vector_type(4)));
typedef unsigned int u32x4 __attribute__((ext_vector_type(4)));
typedef unsigned int u32x8 __attribute__((ext_vector_type(8)));
}

// Low 32 bits of a generic pointer to LDS == LDS byte address (ISA 10.2 aperture rule).
__device__ __forceinline__ unsigned lds_off(const void* p) {
  return (unsigned)(unsigned long long)(size_t)p;
}

// One TDM descriptor: 1-D copy of kTileElems f32 (32 KB, contiguous) -> LDS.
// Group0: count=1 | lds_addr | global_addr[56:0] | type=2 ("image").
// Group1: data_size=4B, tensor_dim0=0xFFFFFFFF (no OOB), tile_dim0=kTileElems,
//         tile_dim1/2 = 0 (1-D), strides don't-care for 1-D.
__device__ __forceinline__ void tdm_issue_tile(const float* gsrc, unsigned lds_base) {
  unsigned long long ga = (unsigned long long)(size_t)gsrc;
  u32x4 g0;
  g0.x = 1u;                                    // count=1 (valid user descriptor)
  g0.y = lds_base;                              // lds_addr (bytes)
  g0.z = (unsigned)ga;                          // global_addr[31:0]
  g0.w = (0x2u << 30) | (unsigned)(ga >> 32);   // type=2 | global_addr[56:32]
  u32x8 g1;
  g1.s0 = 0x00020000u;                          // data_size=2 (4B), wg_mask=0
  g1.s1 = 0xFFFF0000u;                          // tensor_dim0[15:0]=0xFFFF
  g1.s2 = 0x0000FFFFu;                          // tensor_dim0[31:16]=0xFFFF, tensor_dim1=0
  g1.s3 = ((unsigned)kTileElems) << 16;         // tile_dim0 = 8192 elements
  g1.s4 = 0u;                                   // tile_dim1=0, tile_dim2=0 (1-D)
  g1.s5 = (unsigned)kTileElems;                 // tensor_dim0_stride (unused for 1-D)
  g1.s6 = 0u;
  g1.s7 = 0u;
  asm volatile("tensor_load_to_lds %0, %1" :: "s"(g0), "s"(g1) : "memory");
}

// Wait until at most k tensor ops remain outstanding (immediate-only ladder).
__device__ __forceinline__ void wait_tensor_le(int k) {
  if (k >= kDepth - 1)  __builtin_amdgcn_s_wait_tensorcnt(kDepth - 1);
  else if (k == 3)      __builtin_amdgcn_s_wait_tensorcnt(3);
  else if (k == 2)      __builtin_amdgcn_s_wait_tensorcnt(2);
  else if (k == 1)      __builtin_amdgcn_s_wait_tensorcnt(1);
  else                  __builtin_amdgcn_s_wait_tensorcnt(0);
}

__global__ __launch_bounds__(kThreads)
void ema_scan_kernel(const float* __restrict__ x, float* __restrict__ y) {
  __shared__ __align__(16) float smem[kNBuf][kTT * kF];

  const int tid   = (int)threadIdx.x;
  const int chunk = (int)blockIdx.x;
  const int b     = (int)blockIdx.y;
  const int t0    = chunk * kChunk;
  const int tbeg  = (chunk == 0) ? 0 : (t0 - kWarm);
  const int ntiles = (t0 + kChunk - tbeg) / kTT;   // 32 (chunk 0) or 48
  const bool issuer = (tid < 32);                  // wave 0 drives the TDM

  const float* xb = x + (size_t)b * kT * kF;
  float*       yb = y + (size_t)b * kT * kF;

  unsigned lbase[kNBuf];
#pragma unroll
  for (int i = 0; i < kNBuf; ++i) lbase[i] = lds_off(&smem[i][0]);

  // Prologue: fill the TDM pipeline with kDepth tiles (wave 0 only; EXEC ignored).
  if (issuer) {
#pragma unroll
    for (int i = 0; i < kDepth; ++i)
      if (i < ntiles)
        tdm_issue_tile(xb + (size_t)(tbeg + i * kTT) * kF, lbase[i % kNBuf]);
  }

  f4 ycar;
  ycar.x = 0.f; ycar.y = 0.f; ycar.z = 0.f; ycar.w = 0.f;

  for (int i = 0; i < ntiles; ++i) {
    const int rem = ntiles - 1 - i;
    // Tensor ops complete in order per wave: leaving <= min(kDepth-1, rem)
    // outstanding guarantees tile i has fully landed in LDS.
    if (issuer) wait_tensor_le(rem < kDepth - 1 ? rem : kDepth - 1);
    __syncthreads();  // publish tile i to all waves; tile i-1 reads all done

    // Refill the ring slot that held tile i-1 (reads finished before barrier).
    if (issuer && (i + kDepth) < ntiles)
      tdm_issue_tile(xb + (size_t)(tbeg + (i + kDepth) * kTT) * kF,
                     lbase[(i + kDepth) % kNBuf]);

    const f4* lsrc = (const f4*)&smem[i % kNBuf][0];
    const int trow = tbeg + i * kTT;
#pragma unroll
    for (int r = 0; r < kTT; ++r) {
      f4 xv = lsrc[r * (kF / 4) + tid];        // ds_load_b128, conflict-free
      const int gt = trow + r;                 // uniform across block
      if (gt == 0) {
        ycar = xv;                             // y_0 = x_0
      } else {
        ycar.x = fmaf(kBeta, ycar.x, kAlpha * xv.x);
        ycar.y = fmaf(kBeta, ycar.y, kAlpha * xv.y);
        ycar.z = fmaf(kBeta, ycar.z, kAlpha * xv.z);
        ycar.w = fmaf(kBeta, ycar.w, kAlpha * xv.w);
      }
      if (gt >= t0) {                          // skip warmup region
        __builtin_nontemporal_store(ycar, (f4*)(yb + (size_t)gt * kF) + tid);
      }
    }
  }
}

extern "C" void kernel_launch(void* const* d_in, const int* in_sizes, int n_in,
                              void* d_out, int out_size, void* d_ws, size_t ws_size,
                              hipStream_t stream) {
  (void)in_sizes; (void)n_in; (void)out_size; (void)d_ws; (void)ws_size;
  const float* x = (const float*)d_in[0];
  float*       y = (float*)d_out;
  dim3 grid(kT / kChunk, kB, 1);   // 16 chunks x 8 batches = 128 blocks (4 waves each)
  dim3 block(kThreads, 1, 1);
  hipLaunchKernelGGL(ema_scan_kernel, grid, block, 0, stream, x, y);
}